// FixedPCA_71201967833960
// MI455X (gfx1250) — compile-verified
//
#include <hip/hip_runtime.h>

typedef __attribute__((ext_vector_type(2))) float v2f;
typedef __attribute__((ext_vector_type(8))) float v8f;

#define HH 512
#define WW 512
#define CIN 3
#define JJ 16
#define HT 16          // output rows per block
#define WT 128         // output cols per block (8 waves * 16)
#define XT_H (HT + 4)  // 20 input rows incl. halo
#define XT_W (WT + 4)  // 132 input cols incl. halo
#define KP 76          // K padded from 75 to multiple of 4
#define NSTEP (KP / 4) // 19 wmma steps

#if __has_builtin(__builtin_amdgcn_global_load_async_to_lds_b32) && \
    __has_builtin(__builtin_amdgcn_s_wait_asynccnt)
#define USE_ASYNC_LDS 1
#endif

// ---------------------------------------------------------------------------
// Kernel 1: per-batch bias = m[b] * coef_sum[j]
// m[b] = sum_{c,y,x} x * cnt(y)*cnt(x) / (75*H*W), cnt in {3,4,5} at borders.
// ---------------------------------------------------------------------------
__global__ __launch_bounds__(256) void fixedpca_bias_kernel(
    const float* __restrict__ x, const float* __restrict__ coef,
    float* __restrict__ bias) {
  __shared__ float red[256];
  __shared__ float msh;
  const int b = blockIdx.x;
  const int tid = threadIdx.x;
  const float* xb = x + (size_t)b * CIN * HH * WW;
  float acc = 0.0f;
  for (int idx = tid; idx < CIN * HH * WW; idx += 256) {
    int sp = idx & (HH * WW - 1);
    int y = sp >> 9;
    int xx = sp & 511;
    int cy = ((y + 2 < 4) ? (y + 2) : 4) - ((y - 509 > 0) ? (y - 509) : 0) + 1;
    int cx = ((xx + 2 < 4) ? (xx + 2) : 4) - ((xx - 509 > 0) ? (xx - 509) : 0) + 1;
    acc += xb[idx] * (float)(cy * cx);
  }
  red[tid] = acc;
  __syncthreads();
  for (int s = 128; s > 0; s >>= 1) {
    if (tid < s) red[tid] += red[tid + s];
    __syncthreads();
  }
  if (tid == 0) msh = red[0] * (1.0f / (75.0f * (float)HH * (float)WW));
  __syncthreads();
  if (tid < JJ) {
    float s = 0.0f;
    for (int i = 0; i < 75; ++i) s += coef[i * JJ + tid];
    bias[b * JJ + tid] = msh * s;
  }
}

// ---------------------------------------------------------------------------
// Kernel 2: implicit-GEMM conv via V_WMMA_F32_16X16X4_F32.
// Per wave: 16x16 (w x j) tile, K = 75 (c-major permuted) padded to 76.
// Input tile staged to LDS via GLOBAL_LOAD_ASYNC_TO_LDS when available.
// ---------------------------------------------------------------------------
__global__ __launch_bounds__(256) void fixedpca_conv_wmma_kernel(
    const float* __restrict__ x, const float* __restrict__ coef,
    const float* __restrict__ bias, float* __restrict__ out) {
  __shared__ float xt[CIN * XT_H * XT_W];  // input tile + halo (zero-filled)
  __shared__ float cp[KP * JJ];            // permuted, zero-padded coef

  const int blk = blockIdx.x;
  const int b = blk >> 7;                  // 32*4 = 128 blocks per batch
  const int h0 = ((blk >> 2) & 31) << 4;   // 32 h-tiles of 16
  const int w0 = (blk & 3) << 7;           // 4 w-tiles of 128
  const int tid = threadIdx.x;

  // Stage coef into LDS with K permuted to c-major: i' = c*25 + ki*5 + kj.
  for (int idx = tid; idx < KP * JJ; idx += 256) {
    int ip = idx >> 4;
    int j = idx & 15;
    float v = 0.0f;
    if (ip < 75) {
      int c = ip / 25;
      int r = ip % 25;
      int ki = r / 5;
      int kj = r % 5;
      v = coef[(ki * 15 + kj * 3 + c) * JJ + j];  // original i = ki*15+kj*3+c
    }
    cp[idx] = v;
  }

  // Stage input tile (rows h0-2..h0+17, cols w0-2..w0+129) with zero halo.
  const float* xb = x + (size_t)b * CIN * HH * WW;
  for (int idx = tid; idx < CIN * XT_H * XT_W; idx += 256) {
    int c = idx / (XT_H * XT_W);
    int r = idx % (XT_H * XT_W);
    int hr = r / XT_W;
    int wc = r % XT_W;
    int y = h0 + hr - 2;
    int xx = w0 + wc - 2;
    bool inb = ((unsigned)y < HH) && ((unsigned)xx < WW);
#ifdef USE_ASYNC_LDS
    typedef __attribute__((address_space(1))) int GI;
    typedef __attribute__((address_space(3))) int LI;
    if (inb) {
      // Async DMA global -> LDS (no VGPR round-trip), tracked by ASYNCcnt.
      __builtin_amdgcn_global_load_async_to_lds_b32(
          (GI*)(xb + ((size_t)c * HH + y) * WW + xx), (LI*)(xt + idx), 0, 0);
    } else {
      xt[idx] = 0.0f;
    }
#else
    float v = 0.0f;
    if (inb) v = xb[((size_t)c * HH + y) * WW + xx];
    xt[idx] = v;
#endif
  }
#ifdef USE_ASYNC_LDS
  __builtin_amdgcn_s_wait_asynccnt(0);
#endif
  __syncthreads();

  const int lane = tid & 31;
  const int wave = tid >> 5;
  const int m = lane & 15;            // M index (w offset) for A; N (j) for B/C
  const int koff = (lane >> 4) << 1;  // half-wave K offset: 0 or 2
  const int wt0 = wave << 4;          // this wave's w-tile offset in block

  // B fragments: b.x = cp[k+koff][n], b.y = cp[k+koff+1][n] (rows k, k+2 per
  // VGPR across half-waves, mirroring the C/D striping).
  v2f bf[NSTEP];
#pragma unroll
  for (int s = 0; s < NSTEP; ++s) {
    int k = 4 * s + koff;
    bf[s].x = cp[k * JJ + m];
    bf[s].y = cp[(k + 1) * JJ + m];
  }
  const float biasv = bias[b * JJ + m];

  // Per-lane h-invariant LDS offsets for the two A elements of each step.
  // K index ip -> (c, ki, kj); clamp padded ip to 74 (B side is zero there).
  int aoff0[NSTEP], aoff1[NSTEP];
#pragma unroll
  for (int s = 0; s < NSTEP; ++s) {
    int ip0 = 4 * s + koff;
    if (ip0 > 74) ip0 = 74;
    int ip1 = ip0 + 1;
    if (ip1 > 74) ip1 = 74;
    int c0 = ip0 / 25, r0 = ip0 % 25;
    int c1 = ip1 / 25, r1 = ip1 % 25;
    aoff0[s] = (c0 * XT_H + (r0 / 5)) * XT_W + wt0 + m + (r0 % 5);
    aoff1[s] = (c1 * XT_H + (r1 / 5)) * XT_W + wt0 + m + (r1 % 5);
  }

  for (int hl = 0; hl < HT; ++hl) {
    const int hofs = hl * XT_W;
    v8f cacc = {};
#pragma unroll
    for (int s = 0; s < NSTEP; ++s) {
      v2f a;
      a.x = xt[aoff0[s] + hofs];
      a.y = xt[aoff1[s] + hofs];
      cacc = __builtin_amdgcn_wmma_f32_16x16x4_f32(
          /*neg_a=*/false, a, /*neg_b=*/false, bf[s],
          /*c_mod=*/(short)0, cacc, /*reuse_a=*/false, /*reuse_b=*/false);
    }
    // C/D layout: lane<16 holds (M=0..7, N=lane); lane>=16 holds (M=8..15,
    // N=lane-16) -> each lane already has 8 contiguous-w values for one j.
    const int j = m;
    const int wseg = (lane < 16) ? 0 : 8;
    size_t o = (((size_t)b * JJ + j) * HH + (h0 + hl)) * WW + w0 + wt0 + wseg;
    float4 v0 = make_float4(cacc[0] - biasv, cacc[1] - biasv,
                            cacc[2] - biasv, cacc[3] - biasv);
    float4 v1 = make_float4(cacc[4] - biasv, cacc[5] - biasv,
                            cacc[6] - biasv, cacc[7] - biasv);
    *(float4*)(out + o) = v0;
    *(float4*)(out + o + 4) = v1;
  }
}

extern "C" void kernel_launch(void* const* d_in, const int* in_sizes, int n_in,
                              void* d_out, int out_size, void* d_ws,
                              size_t ws_size, hipStream_t stream) {
  (void)in_sizes; (void)n_in; (void)out_size; (void)ws_size;
  const float* x = (const float*)d_in[0];     // [32,3,512,512]
  const float* coef = (const float*)d_in[1];  // [75,16]
  float* out = (float*)d_out;                 // [32,16,512,512]
  float* bias = (float*)d_ws;                 // [32,16] scratch

  fixedpca_bias_kernel<<<32, 256, 0, stream>>>(x, coef, bias);
  const int nblk = 32 * (HH / HT) * (WW / WT);  // 32*32*4 = 4096
  fixedpca_conv_wmma_kernel<<<nblk, 256, 0, stream>>>(x, coef, bias, out);
}